// GCN_64974265253907
// MI455X (gfx1250) — compile-verified
//
#include <hip/hip_runtime.h>

typedef __attribute__((ext_vector_type(2))) float v2f;
typedef __attribute__((ext_vector_type(8))) float v8f;

#define HIDDEN 128

// ---------------- in-degree (f32 so it can be used directly as divisor) ----
__global__ void degree_kernel(const int* __restrict__ dst, float* __restrict__ cnt, int E) {
    int e = blockIdx.x * blockDim.x + threadIdx.x;
    if (e < E) atomicAdd(&cnt[dst[e]], 1.0f);
}

// ---------------- edge gather + scatter-add: agg[dst] += h[src] ------------
// One thread per (edge, 4-dim chunk). float4 gather, 4x global_atomic_add_f32.
// Working set (h: 51MB, agg: 51MB) is L2-resident on MI455X (192MB L2).
__global__ void edge_scatter(const int* __restrict__ src, const int* __restrict__ dst,
                             const float* __restrict__ h, float* __restrict__ agg,
                             int E, int D) {
    int t = blockIdx.x * blockDim.x + threadIdx.x;
    int chunks = D >> 2;              // D is 64 or 128 (power of two)
    int e = t / chunks;
    if (e >= E) return;
    int c = t - e * chunks;
    int s = src[e], d = dst[e];
    const float4 v = ((const float4*)(h + (size_t)s * D))[c];
    float* o = agg + (size_t)d * D + (c << 2);
    atomicAdd(o + 0, v.x);
    atomicAdd(o + 1, v.y);
    atomicAdd(o + 2, v.z);
    atomicAdd(o + 3, v.w);
}

// ---------------- WMMA SAGE GEMM: out = relu((agg/deg)@Wl + hin@Wr + b) ----
// Register-tiled: one wave computes a 32x64 output block = 2 row-tiles x
// 4 col-tiles = 8 independent v8f accumulators. Per K-step (K+=4):
//   2 A b64 loads + 8 B b32 loads feed 8 V_WMMA_F32_16X16X4_F32 ops
// (1.25 loads/WMMA), and the 8 WMMAs write distinct accumulators so there
// are no WMMA->WMMA RAW hazards (ISA 7.12.1) -- the XDL pipe stays full.
// A frag (16x4 f32): lanes 0-15 -> M=lane, {K,K+1}; lanes 16-31 -> {K+2,K+3}.
// B frag (4x16 f32): lanes 0-15 -> N=lane, rows {K,K+1}; lanes 16-31 -> {K+2,K+3}.
// D (16x16 f32): VGPR v, lanes 0-15 -> M=v, N=lane; lanes 16-31 -> M=v+8.
// nrows must be a multiple of 32 (N=100000 is) so EXEC stays all-1s per wave.
__global__ void sage_gemm_relu(const float* __restrict__ agg,
                               const float* __restrict__ hin,
                               const float* __restrict__ cnt,
                               const float* __restrict__ Wl,
                               const float* __restrict__ Wr,
                               const float* __restrict__ bias,
                               float* __restrict__ hout,
                               int nrows, int din) {
    const int lane = threadIdx.x & 31;
    const int gw   = blockIdx.x * (blockDim.x >> 5) + (threadIdx.x >> 5);
    const int ncg  = HIDDEN / 64;                 // 2 column groups of 64
    const int rg   = gw / ncg;                    // row group (32 rows)
    const int cg   = gw - rg * ncg;
    const int row0 = rg * 32;
    if (row0 >= nrows) return;                    // wave-uniform

    const int half = lane >> 4;                   // 0 or 1
    const int l    = lane & 15;
    const int ar0  = row0 + l;                    // A row, tile 0
    const int ar1  = row0 + 16 + l;               // A row, tile 1
    const int bc   = cg * 64 + l;                 // B col base (tiles at +0,+16,+32,+48)
    const float inv0 = 1.0f / fmaxf(cnt[ar0], 1.0f);
    const float inv1 = 1.0f / fmaxf(cnt[ar1], 1.0f);

    v8f acc[8] = {};                              // acc[rt*4 + j]

    // Part 1: (agg * 1/deg) @ Wl
    const float* A0 = agg + (size_t)ar0 * din;
    const float* A1 = agg + (size_t)ar1 * din;
    #pragma unroll 2
    for (int k = 0; k < din; k += 4) {
        const int kk = k + 2 * half;
        v2f a0 = *(const v2f*)(A0 + kk); a0 *= inv0;
        v2f a1 = *(const v2f*)(A1 + kk); a1 *= inv1;
        const float* w0 = Wl + (size_t)kk * HIDDEN + bc;
        const float* w1 = Wl + (size_t)(kk + 1) * HIDDEN + bc;
        #pragma unroll
        for (int j = 0; j < 4; j++) {
            v2f b;
            b.x = w0[16 * j];
            b.y = w1[16 * j];
            acc[j] = __builtin_amdgcn_wmma_f32_16x16x4_f32(
                false, a0, false, b, (short)0, acc[j], false, false);
            acc[4 + j] = __builtin_amdgcn_wmma_f32_16x16x4_f32(
                false, a1, false, b, (short)0, acc[4 + j], false, false);
        }
    }
    // Part 2: hin @ Wr
    const float* H0 = hin + (size_t)ar0 * din;
    const float* H1 = hin + (size_t)ar1 * din;
    #pragma unroll 2
    for (int k = 0; k < din; k += 4) {
        const int kk = k + 2 * half;
        v2f a0 = *(const v2f*)(H0 + kk);
        v2f a1 = *(const v2f*)(H1 + kk);
        const float* w0 = Wr + (size_t)kk * HIDDEN + bc;
        const float* w1 = Wr + (size_t)(kk + 1) * HIDDEN + bc;
        #pragma unroll
        for (int j = 0; j < 4; j++) {
            v2f b;
            b.x = w0[16 * j];
            b.y = w1[16 * j];
            acc[j] = __builtin_amdgcn_wmma_f32_16x16x4_f32(
                false, a0, false, b, (short)0, acc[j], false, false);
            acc[4 + j] = __builtin_amdgcn_wmma_f32_16x16x4_f32(
                false, a1, false, b, (short)0, acc[4 + j], false, false);
        }
    }

    // Epilogue: bias + ReLU + store
    #pragma unroll
    for (int rt = 0; rt < 2; rt++) {
        #pragma unroll
        for (int j = 0; j < 4; j++) {
            const int col = bc + 16 * j;
            const float bv = bias[col];
            #pragma unroll
            for (int v = 0; v < 8; v++) {
                const int orow = row0 + rt * 16 + v + 8 * half;
                hout[(size_t)orow * HIDDEN + col] = fmaxf(acc[rt * 4 + j][v] + bv, 0.0f);
            }
        }
    }
}

// ---------------- global mean pool (sum + count; divide later) -------------
__global__ void pool_kernel(const float* __restrict__ h, const int* __restrict__ batch,
                            float* __restrict__ gsum, float* __restrict__ gcnt, int n) {
    long long t = (long long)blockIdx.x * blockDim.x + threadIdx.x;
    int node = (int)(t >> 7);                     // D = 128
    if (node >= n) return;
    int d = (int)(t & 127);
    int g = batch[node];
    atomicAdd(&gsum[(size_t)g * HIDDEN + d], h[(size_t)node * HIDDEN + d]);
    if (d == 0) atomicAdd(&gcnt[g], 1.0f);
}

// ---------------- mixture prefix offsets + contrib (tiny, serial) ----------
__global__ void mix_setup(const int* __restrict__ sizes, const float* __restrict__ fracs,
                          int* __restrict__ offsets, float* __restrict__ contrib, int M) {
    if (blockIdx.x == 0 && threadIdx.x == 0) {
        int off = 0;
        for (int m = 0; m < M; m++) {
            offsets[m] = off;
            float s = 0.0f;
            int sz = sizes[m];
            for (int i = 0; i < sz; i++) s += fracs[off + i];
            contrib[m] = s;
            off += sz;
        }
        offsets[M] = off;
    }
}

// ---------------- fused mixture weighted-sum + FC (one wave per mixture) ---
__global__ void mix_fc(const float* __restrict__ gsum, const float* __restrict__ gcnt,
                       const float* __restrict__ fracs, const int* __restrict__ offsets,
                       const float* __restrict__ contrib, const float* __restrict__ fcW,
                       const float* __restrict__ fcb, float* __restrict__ out, int M) {
    int wave = (blockIdx.x * blockDim.x + threadIdx.x) >> 5;
    int lane = threadIdx.x & 31;
    if (wave >= M) return;
    int g0 = offsets[wave], g1 = offsets[wave + 1];
    float invc = 1.0f / contrib[wave];            // fracs >= 0.05 so contrib > 0
    float acc0 = 0.f, acc1 = 0.f, acc2 = 0.f, acc3 = 0.f;
    for (int g = g0; g < g1; g++) {
        float w = fracs[g] * invc / fmaxf(gcnt[g], 1.0f);
        const float* row = gsum + (size_t)g * HIDDEN;
        acc0 += row[lane +  0] * w;
        acc1 += row[lane + 32] * w;
        acc2 += row[lane + 64] * w;
        acc3 += row[lane + 96] * w;
    }
    float s = acc0 * fcW[lane] + acc1 * fcW[lane + 32] +
              acc2 * fcW[lane + 64] + acc3 * fcW[lane + 96];
    for (int m = 16; m > 0; m >>= 1) s += __shfl_xor(s, m, 32);
    if (lane == 0) out[wave] = s + fcb[0];
}

extern "C" void kernel_launch(void* const* d_in, const int* in_sizes, int n_in,
                              void* d_out, int out_size, void* d_ws, size_t ws_size,
                              hipStream_t stream) {
    const float* x          = (const float*)d_in[0];
    const int*   edge_index = (const int*)d_in[1];
    const int*   batch      = (const int*)d_in[2];
    const int*   mix_sizes  = (const int*)d_in[3];
    const float* fracs      = (const float*)d_in[4];
    const float* W1l = (const float*)d_in[5];
    const float* W1r = (const float*)d_in[6];
    const float* b1  = (const float*)d_in[7];
    const float* W2l = (const float*)d_in[8];
    const float* W2r = (const float*)d_in[9];
    const float* b2  = (const float*)d_in[10];
    const float* fcW = (const float*)d_in[11];
    const float* fcb = (const float*)d_in[12];
    float* out = (float*)d_out;

    const int IN_DIM = 64;
    const int N = in_sizes[0] / IN_DIM;
    const int E = in_sizes[1] / 2;
    const int M = in_sizes[3];
    const int G = in_sizes[4];
    const int* src = edge_index;
    const int* dst = edge_index + E;

    // Workspace layout (floats): agg | hA | hB | cnt | gsum | gcnt | contrib | offsets(int)
    float* ws      = (float*)d_ws;
    float* agg     = ws;
    float* hA      = agg  + (size_t)N * HIDDEN;
    float* hB      = hA   + (size_t)N * HIDDEN;
    float* cnt     = hB   + (size_t)N * HIDDEN;
    float* gsum    = cnt  + N;
    float* gcnt    = gsum + (size_t)G * HIDDEN;
    float* contrib = gcnt + G;
    int*   offsets = (int*)(contrib + M);

    // In-degree (shared across all layers)
    hipMemsetAsync(cnt, 0, sizeof(float) * (size_t)N, stream);
    degree_kernel<<<(E + 255) / 256, 256, 0, stream>>>(dst, cnt, E);

    // One wave per 32x64 output block
    const int gemm_waves  = (N / 32) * (HIDDEN / 64);
    const int gemm_blocks = (gemm_waves + 7) / 8;

    // Layer 1: 64 -> 128
    hipMemsetAsync(agg, 0, sizeof(float) * (size_t)N * IN_DIM, stream);
    {
        long long T = (long long)E * (IN_DIM / 4);
        edge_scatter<<<(int)((T + 255) / 256), 256, 0, stream>>>(src, dst, x, agg, E, IN_DIM);
    }
    sage_gemm_relu<<<gemm_blocks, 256, 0, stream>>>(agg, x, cnt, W1l, W1r, b1, hA, N, IN_DIM);

    // Layers 2..4: 128 -> 128 (conv2 reused 3 times)
    float* hcur = hA;
    float* hnext = hB;
    for (int layer = 0; layer < 3; layer++) {
        hipMemsetAsync(agg, 0, sizeof(float) * (size_t)N * HIDDEN, stream);
        long long T = (long long)E * (HIDDEN / 4);
        edge_scatter<<<(int)((T + 255) / 256), 256, 0, stream>>>(src, dst, hcur, agg, E, HIDDEN);
        sage_gemm_relu<<<gemm_blocks, 256, 0, stream>>>(agg, hcur, cnt, W2l, W2r, b2, hnext, N, HIDDEN);
        float* t = hcur; hcur = hnext; hnext = t;
    }

    // Global mean pool
    hipMemsetAsync(gsum, 0, sizeof(float) * (size_t)G * HIDDEN, stream);
    hipMemsetAsync(gcnt, 0, sizeof(float) * (size_t)G, stream);
    {
        long long T = (long long)N * HIDDEN;
        pool_kernel<<<(int)((T + 255) / 256), 256, 0, stream>>>(hcur, batch, gsum, gcnt, N);
    }

    // Mixture scatter + FC
    mix_setup<<<1, 1, 0, stream>>>(mix_sizes, fracs, offsets, contrib, M);
    mix_fc<<<(M * 32 + 255) / 256, 256, 0, stream>>>(gsum, gcnt, fracs, offsets, contrib,
                                                     fcW, fcb, out, M);
}